// MotionDeformableAttention_13769665150988
// MI455X (gfx1250) — compile-verified
//
#include <hip/hip_runtime.h>
#include <math.h>

typedef __attribute__((ext_vector_type(2))) float v2f;
typedef __attribute__((ext_vector_type(8))) float v8f;

// ---------------------------------------------------------------------------
// fp32 WMMA GEMM:  C[b][m][n] = sum_k A[b][m][k] * W[k][n] + bias[n],  K = 256
//   Block = 256 threads (8 waves). Block computes a 64-row stripe of C.
//   A tile (64x256) staged in LDS (padded stride 260 -> conflict-free).
//   Wave w owns columns [(blockIdx.y*8+w)*16, +16) and keeps FOUR 16x16
//   accumulators (rows m0+0..15, +16..31, +32..47, +48..63), so each B
//   fragment load is reused by 4 v_wmma_f32_16x16x4_f32 ops.
// ---------------------------------------------------------------------------
__global__ void __launch_bounds__(256)
gemm_f32_wmma(const float* __restrict__ A, long long lda, long long abatch,
              const float* __restrict__ W, int N,
              const float* __restrict__ bias,
              float* __restrict__ C, long long cbatch,
              int M)
{
    __shared__ float As[64][260];

    const int b    = blockIdx.z;
    const int m0   = blockIdx.x << 6;      // 64-row stripe
    const int tid  = threadIdx.x;
    const int wave = tid >> 5;
    const int lane = tid & 31;
    const int half = lane >> 4;            // 0: lanes 0-15, 1: lanes 16-31
    const int lid  = lane & 15;
    const int n0   = ((blockIdx.y << 3) + wave) << 4;

    const float* Ab = A + (long long)b * abatch;

    // Cooperative, coalesced load of the 64x256 A tile (64 floats/thread).
#pragma unroll
    for (int i = 0; i < 64; ++i) {
        int idx = tid + (i << 8);          // 0..16383
        int r = idx >> 8;
        int c = idx & 255;
        As[r][c] = Ab[(long long)(m0 + r) * lda + c];
    }
    __syncthreads();

    if (n0 < N) {                          // wave-uniform: EXEC stays all-1s
        v8f acc0 = {}, acc1 = {}, acc2 = {}, acc3 = {};
        const float* Wc = W + n0 + lid;    // column n0+lid of W (row-major KxN)
        __builtin_prefetch(Wc, 0, 1);      // global_prefetch_b8 (L2 warm-up)
#pragma unroll 4
        for (int k = 0; k < 256; k += 4) {
            // B fragment: element (k = 2*half + j, n = lid) in vgpr j
            v2f bb;
            bb.x = Wc[(long long)(k + 2 * half) * N];
            bb.y = Wc[(long long)(k + 2 * half + 1) * N];
            // A fragments: element (m = lid, k = 2*half + j) in vgpr j
            v2f a0, a1, a2, a3;
            a0.x = As[lid     ][k + 2 * half];
            a0.y = As[lid     ][k + 2 * half + 1];
            a1.x = As[lid + 16][k + 2 * half];
            a1.y = As[lid + 16][k + 2 * half + 1];
            a2.x = As[lid + 32][k + 2 * half];
            a2.y = As[lid + 32][k + 2 * half + 1];
            a3.x = As[lid + 48][k + 2 * half];
            a3.y = As[lid + 48][k + 2 * half + 1];
            acc0 = __builtin_amdgcn_wmma_f32_16x16x4_f32(
                false, a0, false, bb, (short)0, acc0, false, false);
            acc1 = __builtin_amdgcn_wmma_f32_16x16x4_f32(
                false, a1, false, bb, (short)0, acc1, false, false);
            acc2 = __builtin_amdgcn_wmma_f32_16x16x4_f32(
                false, a2, false, bb, (short)0, acc2, false, false);
            acc3 = __builtin_amdgcn_wmma_f32_16x16x4_f32(
                false, a3, false, bb, (short)0, acc3, false, false);
        }
        const float bv = bias ? bias[n0 + lid] : 0.0f;
        float* Cb = C + (long long)b * cbatch + n0 + lid;
#pragma unroll
        for (int j = 0; j < 8; ++j) {
            int m = (half << 3) + j;        // C: (m = 8*half + j, n = lid)
            Cb[(long long)(m0 + m)      * N] = acc0[j] + bv;
            Cb[(long long)(m0 + m + 16) * N] = acc1[j] + bv;
            Cb[(long long)(m0 + m + 32) * N] = acc2[j] + bv;
            Cb[(long long)(m0 + m + 48) * N] = acc3[j] + bv;
        }
    }
}

// ---------------------------------------------------------------------------
// Deformable sampling: one wave per (b, q, h); lane = head-dim channel (HD=32)
// ---------------------------------------------------------------------------
__global__ void __launch_bounds__(256)
md_sample_kernel(const float* __restrict__ vproj,   // (bs, nv, 256)
                 const float* __restrict__ off,     // (bs*nq, 256)
                 const float* __restrict__ attnlog, // (bs*nq, 128)
                 const float* __restrict__ trajs,   // (bs, 256, 6, 12, 4, 2)
                 const float* __restrict__ centers, // (bs, 256, 2)
                 float* __restrict__ sampout)       // (bs*nq, 256)
{
    const int NV = 21760;
    const int wid  = (blockIdx.x << 3) + (threadIdx.x >> 5);
    const int lane = threadIdx.x & 31;

    const int h  = wid & 7;
    const int bq = wid >> 3;        // b*1536 + q
    const int q  = bq % 1536;
    const int b  = bq / 1536;
    const int a  = q / 6;
    const int m  = q % 6;

    // ---- softmax over 16 (NL*NP) logits for this head --------------------
    const float* lg = attnlog + (long long)bq * 128 + h * 16;
    float w16[16];
    float mx = -3.0e38f;
#pragma unroll
    for (int i = 0; i < 16; ++i) { w16[i] = lg[i]; mx = fmaxf(mx, w16[i]); }
    float sum = 0.0f;
#pragma unroll
    for (int i = 0; i < 16; ++i) { w16[i] = __expf(w16[i] - mx); sum += w16[i]; }
    const float inv = 1.0f / sum;

    // ---- reference point (last trajectory step + detection center) ------
    const float* ofb = off + (long long)bq * 256 + h * 32;
    const float* tr  = trajs + ((((long long)(b * 256 + a) * 6 + m) * 12 + 11) * 4) * 2;
    const float cx = centers[(b * 256 + a) * 2 + 0];
    const float cy = centers[(b * 256 + a) * 2 + 1];

    const int Hs[4]  = {128, 64, 32, 16};
    const int Wss[4] = {128, 64, 32, 16};
    const int st[4]  = {0, 16384, 20480, 21504};

    const float* vb = vproj + ((long long)b * NV) * 256 + h * 32 + lane;

    float acc = 0.0f;
#pragma unroll
    for (int lv = 0; lv < 4; ++lv) {
        const float Wf = (float)Wss[lv];
        const float Hf = (float)Hs[lv];
        const float rx = (tr[lv * 2 + 0] + cx + 51.2f) * (1.0f / 102.4f);
        const float ry = (tr[lv * 2 + 1] + cy + 51.2f) * (1.0f / 102.4f);
#pragma unroll
        for (int p = 0; p < 4; ++p) {
            const float ox = ofb[lv * 8 + p * 2 + 0];
            const float oy = ofb[lv * 8 + p * 2 + 1];
            const float x = (rx + ox / Wf) * Wf - 0.5f;
            const float y = (ry + oy / Hf) * Hf - 0.5f;
            const float x0 = floorf(x), y0 = floorf(y);
            float s = 0.0f;
#pragma unroll
            for (int c = 0; c < 4; ++c) {
                const float xi = x0 + (float)(c & 1);
                const float yi = y0 + (float)(c >> 1);
                const float w  = (1.0f - fabsf(x - xi)) * (1.0f - fabsf(y - yi));
                if (xi >= 0.0f && xi < Wf && yi >= 0.0f && yi < Hf && w != 0.0f) {
                    const int idx = st[lv] + (int)yi * Wss[lv] + (int)xi;
                    s += w * vb[(long long)idx * 256];   // coalesced 128B/wave
                }
            }
            acc += w16[lv * 4 + p] * inv * s;
        }
    }
    sampout[(long long)bq * 256 + h * 32 + lane] = acc;
}

// ---------------------------------------------------------------------------
// LayerNorm(256) + ReLU + residual, one 256-thread block per row
// ---------------------------------------------------------------------------
__global__ void __launch_bounds__(256)
ln_relu_res_kernel(const float* __restrict__ o,
                   const float* __restrict__ query,
                   const float* __restrict__ g,
                   const float* __restrict__ be,
                   float* __restrict__ out)
{
    __shared__ float s[256];
    const int row = blockIdx.x;
    const int tid = threadIdx.x;
    const long long base = (long long)row * 256 + tid;

    const float x = o[base];
    s[tid] = x;
    __syncthreads();
    for (int k = 128; k > 0; k >>= 1) {
        if (tid < k) s[tid] += s[tid + k];
        __syncthreads();
    }
    const float mu = s[0] * (1.0f / 256.0f);
    __syncthreads();

    const float d = x - mu;
    s[tid] = d * d;
    __syncthreads();
    for (int k = 128; k > 0; k >>= 1) {
        if (tid < k) s[tid] += s[tid + k];
        __syncthreads();
    }
    const float var = s[0] * (1.0f / 256.0f);

    float y = d * rsqrtf(var + 1e-5f) * g[tid] + be[tid];
    y = fmaxf(y, 0.0f);
    out[base] = y + query[base];
}

// ---------------------------------------------------------------------------
// Host launcher
// ---------------------------------------------------------------------------
extern "C" void kernel_launch(void* const* d_in, const int* in_sizes, int n_in,
                              void* d_out, int out_size, void* d_ws, size_t ws_size,
                              hipStream_t stream)
{
    const float* query   = (const float*)d_in[0];   // (4,256,6,256)
    const float* value   = (const float*)d_in[1];   // (21760,4,256)
    const float* trajs   = (const float*)d_in[2];   // (4,256,6,12,4,2)
    const float* centers = (const float*)d_in[3];   // (4,256,2)
    // d_in[4] spatial_shapes, d_in[5] level_start_index: compile-time constants
    const float* W_off   = (const float*)d_in[6];   // (256,256)
    const float* b_off   = (const float*)d_in[7];
    const float* W_attn  = (const float*)d_in[8];   // (256,128)
    const float* b_attn  = (const float*)d_in[9];
    const float* W_val   = (const float*)d_in[10];  // (256,256)
    const float* b_val   = (const float*)d_in[11];
    const float* W_out   = (const float*)d_in[12];  // (256,256)
    const float* b_out   = (const float*)d_in[13];
    const float* ln_g    = (const float*)d_in[14];
    const float* ln_b    = (const float*)d_in[15];

    const int NV = 21760, BS = 4, NQ = 1536, E = 256;
    const long long M_q = (long long)BS * NQ;       // 6144

    float* ws    = (float*)d_ws;
    float* vproj = ws;                                   // 4*21760*256
    float* off   = vproj + (long long)BS * NV * E;       // 6144*256
    float* attn  = off   + M_q * E;                      // 6144*128
    float* samp  = attn  + M_q * 128;                    // 6144*256
    float* o     = off;  // reuse: `off` is fully consumed before out-proj

    dim3 blk(256);

    // 1) value projection: per-batch GEMM (nv x 256) @ (256 x 256)
    gemm_f32_wmma<<<dim3(NV / 64, 2, BS), blk, 0, stream>>>(
        value, (long long)BS * E, (long long)E, W_val, E, b_val,
        vproj, (long long)NV * E, NV);

    // 2) sampling offsets: (6144 x 256) @ (256 x 256)
    gemm_f32_wmma<<<dim3((int)(M_q / 64), 2, 1), blk, 0, stream>>>(
        query, (long long)E, 0LL, W_off, E, b_off, off, 0LL, (int)M_q);

    // 3) attention logits: (6144 x 256) @ (256 x 128)
    gemm_f32_wmma<<<dim3((int)(M_q / 64), 1, 1), blk, 0, stream>>>(
        query, (long long)E, 0LL, W_attn, 128, b_attn, attn, 0LL, (int)M_q);

    // 4) softmax + bilinear deformable sampling (one wave per (b,q,h))
    md_sample_kernel<<<dim3((BS * NQ * 8) / 8), blk, 0, stream>>>(
        vproj, off, attn, trajs, centers, samp);

    // 5) output projection: (6144 x 256) @ (256 x 256)
    gemm_f32_wmma<<<dim3((int)(M_q / 64), 2, 1), blk, 0, stream>>>(
        samp, (long long)E, 0LL, W_out, E, b_out, o, 0LL, (int)M_q);

    // 6) LayerNorm + ReLU + residual
    ln_relu_res_kernel<<<dim3((int)M_q), blk, 0, stream>>>(
        o, query, ln_g, ln_b, (float*)d_out);
}